// GATModel_11785390260481
// MI455X (gfx1250) — compile-verified
//
#include <hip/hip_runtime.h>
#include <hip/hip_bf16.h>
#include <math.h>

#define N_NODES 10000
#define N_EDGES 160000
#define IN_DIM  128
#define HID     16
#define HEADS   50
#define OUT_DIM 10
#define N_GRAPHS 8
#define NEG_SLOPE 0.2f

#define HD (HEADS * HID)          // 800
#define E_TOT (N_EDGES + N_NODES) // 170000 (edges + self loops)

typedef __attribute__((ext_vector_type(2))) float v2f;
typedef __attribute__((ext_vector_type(8))) float v8f;

// ---------------------------------------------------------------------------
// Kernel 1: xp = x @ W  via V_WMMA_F32_16X16X4_F32, one wave per 16x16 tile.
// x: [10000,128] row-major, W: [128,800] row-major, xp: [10000,800] row-major.
// A layout: lane l<16 -> (M=l, K=k+0/1), lane l>=16 -> (M=l-16, K=k+2/3)
// B layout (symmetric): lane l<16 -> (K=k+0/1, N=n0+l), upper half K=k+2/3
// D layout: vgpr r -> row r (+8 upper half), col lane&15
// ---------------------------------------------------------------------------
__global__ __launch_bounds__(32) void gat_gemm_xw(const float* __restrict__ x,
                                                  const float* __restrict__ W,
                                                  float* __restrict__ xp) {
    const int row0 = blockIdx.x * 16;
    const int col0 = blockIdx.y * 16;
    const int lane = threadIdx.x;
    const int half = lane >> 4;      // 0 or 1
    const int l    = lane & 15;

    const float* __restrict__ arow = x + (size_t)(row0 + l) * IN_DIM;
    const float* __restrict__ bcol = W + col0 + l;

    v8f acc = {};
#pragma unroll
    for (int k = 0; k < IN_DIM; k += 4) {
        const int ka = k + half * 2;
        v2f a, b;
        a.x = arow[ka];
        a.y = arow[ka + 1];
        b.x = bcol[(size_t)ka * HD];
        b.y = bcol[(size_t)(ka + 1) * HD];
        acc = __builtin_amdgcn_wmma_f32_16x16x4_f32(
            /*neg_a=*/false, a, /*neg_b=*/false, b,
            /*c_mod=*/(short)0, acc, /*reuse_a=*/false, /*reuse_b=*/false);
    }

    float* __restrict__ out = xp + (size_t)row0 * HD + col0;
#pragma unroll
    for (int r = 0; r < 8; ++r) {
        const int m = r + half * 8;
        out[(size_t)m * HD + l] = acc[r];
    }
}

// ---------------------------------------------------------------------------
// Kernel 2: alpha_s[n,h] = xp[n,h,:] . a_src[h,:]; same for alpha_d.
// One thread per (n,h).
// ---------------------------------------------------------------------------
__global__ void gat_alpha(const float* __restrict__ xp,
                          const float* __restrict__ a_src,
                          const float* __restrict__ a_dst,
                          float* __restrict__ as, float* __restrict__ ad) {
    const int i = blockIdx.x * blockDim.x + threadIdx.x;
    if (i >= N_NODES * HEADS) return;
    const int h = i % HEADS;
    const float* __restrict__ v = xp + (size_t)i * HID;
    float s = 0.f, d = 0.f;
#pragma unroll
    for (int c = 0; c < HID; ++c) {
        const float t = v[c];
        s = fmaf(t, a_src[h * HID + c], s);
        d = fmaf(t, a_dst[h * HID + c], d);
    }
    as[i] = s;
    ad[i] = d;
}

// ---------------------------------------------------------------------------
// Kernel 3: init accumulators. m <- -FLT_MAX, (denom,hbuf,gsum,cnt) <- 0.
// ---------------------------------------------------------------------------
#define ZERO_FLOATS (N_NODES * HEADS + N_NODES * HID + N_GRAPHS * HID + N_GRAPHS)
__global__ void gat_init(float* __restrict__ m, float* __restrict__ zero_base) {
    const int i = blockIdx.x * blockDim.x + threadIdx.x;
    if (i < N_NODES * HEADS) m[i] = -3.402823466e38f;
    if (i < ZERO_FLOATS) zero_base[i] = 0.f;
}

__device__ __forceinline__ float leaky(float v) {
    return v > 0.f ? v : NEG_SLOPE * v;
}

__device__ __forceinline__ void edge_endpoints(int e, const int* __restrict__ ei,
                                               int& src, int& dst) {
    if (e < N_EDGES) { src = ei[e]; dst = ei[N_EDGES + e]; }
    else             { src = e - N_EDGES; dst = src; }       // self loop
}

// ---------------------------------------------------------------------------
// Kernel 4: segment max of leakyrelu logits over edges, per (dst, head).
// One thread per (edge, head).
// ---------------------------------------------------------------------------
__global__ void gat_segmax(const int* __restrict__ ei,
                           const float* __restrict__ as,
                           const float* __restrict__ ad,
                           float* __restrict__ m) {
    const int i = blockIdx.x * blockDim.x + threadIdx.x;
    if (i >= E_TOT * HEADS) return;
    const int e = i / HEADS, h = i - e * HEADS;
    int src, dst; edge_endpoints(e, ei, src, dst);
    const float logit = leaky(as[src * HEADS + h] + ad[dst * HEADS + h]);
    atomicMax(&m[dst * HEADS + h], logit);
}

// ---------------------------------------------------------------------------
// Kernel 5: denom[dst,h] += exp(logit - m[dst,h]). One thread per (edge, head).
// ---------------------------------------------------------------------------
__global__ void gat_segsum(const int* __restrict__ ei,
                           const float* __restrict__ as,
                           const float* __restrict__ ad,
                           const float* __restrict__ m,
                           float* __restrict__ denom) {
    const int i = blockIdx.x * blockDim.x + threadIdx.x;
    if (i >= E_TOT * HEADS) return;
    const int e = i / HEADS, h = i - e * HEADS;
    int src, dst; edge_endpoints(e, ei, src, dst);
    const float logit = leaky(as[src * HEADS + h] + ad[dst * HEADS + h]);
    atomicAdd(&denom[dst * HEADS + h], expf(logit - m[dst * HEADS + h]));
}

// ---------------------------------------------------------------------------
// Kernel 6: weighted scatter-aggregate with head-mean folded in.
// One thread per (edge, channel): reduce over 50 heads in registers, then a
// single atomicAdd into hbuf[dst, c]. 16-thread groups share one edge, so the
// per-head scalar loads are uniform and the xp loads are coalesced over c.
// ---------------------------------------------------------------------------
__global__ void gat_aggregate(const int* __restrict__ ei,
                              const float* __restrict__ as,
                              const float* __restrict__ ad,
                              const float* __restrict__ m,
                              const float* __restrict__ denom,
                              const float* __restrict__ xp,
                              float* __restrict__ hbuf) {
    const int i = blockIdx.x * blockDim.x + threadIdx.x;
    if (i >= E_TOT * HID) return;
    const int e = i >> 4, c = i & (HID - 1);
    int src, dst; edge_endpoints(e, ei, src, dst);
    const float* __restrict__ asrc = as + src * HEADS;
    const float* __restrict__ adst = ad + dst * HEADS;
    const float* __restrict__ mdst = m + dst * HEADS;
    const float* __restrict__ ddst = denom + dst * HEADS;
    const float* __restrict__ xsrc = xp + (size_t)src * HD + c;
    float acc = 0.f;
#pragma unroll 5
    for (int h = 0; h < HEADS; ++h) {
        const float logit = leaky(asrc[h] + adst[h]);
        const float w = expf(logit - mdst[h]) / ddst[h];
        acc = fmaf(w, xsrc[(size_t)h * HID], acc);
    }
    atomicAdd(&hbuf[dst * HID + c], acc);
}

// ---------------------------------------------------------------------------
// Kernel 7: node features -> graph pool accumulation.
// h[n,c] = hbuf[n,c]/HEADS + bias[c]; gsum[batch[n],c] += h; cnt[g] += 1.
// ---------------------------------------------------------------------------
__global__ void gat_pool(const float* __restrict__ hbuf,
                         const float* __restrict__ bias,
                         const int* __restrict__ batch,
                         float* __restrict__ gsum, float* __restrict__ cnt) {
    const int i = blockIdx.x * blockDim.x + threadIdx.x;
    if (i >= N_NODES * HID) return;
    const int n = i >> 4, c = i & (HID - 1);
    const float hv = hbuf[i] * (1.0f / HEADS) + bias[c];
    const int g = batch[n];
    atomicAdd(&gsum[g * HID + c], hv);
    if (c == 0) atomicAdd(&cnt[g], 1.0f);
}

// ---------------------------------------------------------------------------
// Kernel 8: out[g,o] = (gsum[g,:]/max(cnt,1)) @ fc_w + fc_b.
// ---------------------------------------------------------------------------
__global__ void gat_fc(const float* __restrict__ gsum, const float* __restrict__ cnt,
                       const float* __restrict__ fc_w, const float* __restrict__ fc_b,
                       float* __restrict__ out) {
    const int i = blockIdx.x * blockDim.x + threadIdx.x;
    if (i >= N_GRAPHS * OUT_DIM) return;
    const int g = i / OUT_DIM, o = i - g * OUT_DIM;
    const float invc = 1.0f / fmaxf(cnt[g], 1.0f);
    float s = fc_b[o];
#pragma unroll
    for (int c = 0; c < HID; ++c)
        s = fmaf(gsum[g * HID + c] * invc, fc_w[c * OUT_DIM + o], s);
    out[i] = s;
}

// ---------------------------------------------------------------------------
extern "C" void kernel_launch(void* const* d_in, const int* in_sizes, int n_in,
                              void* d_out, int out_size, void* d_ws, size_t ws_size,
                              hipStream_t stream) {
    const float* x     = (const float*)d_in[0];   // [10000,128]
    const int*   ei    = (const int*)d_in[1];     // [2,160000]
    const int*   batch = (const int*)d_in[2];     // [10000]
    // d_in[3] = num_graphs scalar (known: 8)
    const float* W     = (const float*)d_in[4];   // [128,800]
    const float* a_src = (const float*)d_in[5];   // [50,16]
    const float* a_dst = (const float*)d_in[6];   // [50,16]
    const float* bias  = (const float*)d_in[7];   // [16]
    const float* fc_w  = (const float*)d_in[8];   // [16,10]
    const float* fc_b  = (const float*)d_in[9];   // [10]
    float* out = (float*)d_out;                   // [8,10]

    // Workspace layout (floats): ~40.6 MB total
    float* ws    = (float*)d_ws;
    float* xp    = ws;                            // N*HD        = 8,000,000
    float* as    = xp + (size_t)N_NODES * HD;     // N*HEADS     =   500,000
    float* ad    = as + N_NODES * HEADS;          //               500,000
    float* m     = ad + N_NODES * HEADS;          //               500,000
    float* denom = m  + N_NODES * HEADS;          //               500,000
    float* hbuf  = denom + N_NODES * HEADS;       // N*HID       =   160,000
    float* gsum  = hbuf + N_NODES * HID;          // G*HID       =       128
    float* cnt   = gsum + N_GRAPHS * HID;         // G           =         8

    // 1) GEMM: one wave per 16x16 tile of xp
    gat_gemm_xw<<<dim3(N_NODES / 16, HD / 16), 32, 0, stream>>>(x, W, xp);

    // 2) per-(node,head) attention coefficients
    {
        const int n = N_NODES * HEADS;
        gat_alpha<<<(n + 255) / 256, 256, 0, stream>>>(xp, a_src, a_dst, as, ad);
    }

    // 3) init accumulators (ws is poisoned by harness)
    {
        const int n = (N_NODES * HEADS > ZERO_FLOATS) ? N_NODES * HEADS : ZERO_FLOATS;
        gat_init<<<(n + 255) / 256, 256, 0, stream>>>(m, denom);
    }

    // 4) segment max over incoming edges
    {
        const long n = (long)E_TOT * HEADS;
        gat_segmax<<<(int)((n + 255) / 256), 256, 0, stream>>>(ei, as, ad, m);
    }

    // 5) softmax denominator
    {
        const long n = (long)E_TOT * HEADS;
        gat_segsum<<<(int)((n + 255) / 256), 256, 0, stream>>>(ei, as, ad, m, denom);
    }

    // 6) weighted aggregate, head-mean folded (one atomic per edge-channel)
    {
        const long n = (long)E_TOT * HID;
        gat_aggregate<<<(int)((n + 255) / 256), 256, 0, stream>>>(ei, as, ad, m, denom,
                                                                  xp, hbuf);
    }

    // 7) graph mean-pool accumulation
    {
        const int n = N_NODES * HID;
        gat_pool<<<(n + 255) / 256, 256, 0, stream>>>(hbuf, bias, batch, gsum, cnt);
    }

    // 8) final FC
    gat_fc<<<1, 128, 0, stream>>>(gsum, cnt, fc_w, fc_b, out);
}